// ParticleFilter_35253091566083
// MI455X (gfx1250) — compile-verified
//
#include <hip/hip_runtime.h>
#include <hip/hip_bf16.h>
#include <math.h>

#define DT 0.1f
#define TILE 1024          // particles per tile (one block's chunk)
#define B1 256             // block size for streaming kernels (8 waves of 32)
#define PT 4               // particles per thread
#define MAX_NT 4096        // tiles for N = 2^22

typedef float v2f __attribute__((ext_vector_type(2)));
typedef float v8f __attribute__((ext_vector_type(8)));

// -------- wave32 sum-reduction on the matrix pipe --------
// A (16x4) holds one f32 per lane (a1 = 0), B = ones(4x16), C = 0.
// D[m][n] = x[m] + x[m+16]  (rowsum, replicated over columns).
// Lane n<16 holds D[0..7][n] in d0..d7; lane n>=16 holds D[8..15][n-16].
// So t = sum(d0..d7) gives rows 0..7 on lanes 0..15 and rows 8..15 on lanes 16..31;
// total = t(lane0) + t(lane16).
__device__ __forceinline__ float wave_reduce_wmma(float x) {
  v2f a; a[0] = x;    a[1] = 0.0f;
  v2f b; b[0] = 1.0f; b[1] = 1.0f;
  v8f c = {0.f, 0.f, 0.f, 0.f, 0.f, 0.f, 0.f, 0.f};
  v8f d = __builtin_amdgcn_wmma_f32_16x16x4_f32(false, a, false, b,
                                                (short)0, c, false, false);
  float t = ((d[0] + d[1]) + (d[2] + d[3])) + ((d[4] + d[5]) + (d[6] + d[7]));
  return __shfl(t, 0, 32) + __shfl(t, 16, 32);
}

// -------- unicycle propagation with colored control noise --------
struct P3 { float x, y, th; };
__device__ __forceinline__ P3 propagate(float px, float py, float th,
                                        float e0, float e1,
                                        float u0, float u1,
                                        float L00, float L10, float L11) {
  float v  = u0 + e0 * L00;
  float om = u1 + e0 * L10 + e1 * L11;
  float sn, cs;
  __sincosf(th, &sn, &cs);
  P3 r;
  r.x  = px + v * DT * cs;
  r.y  = py + v * DT * sn;
  r.th = th + om * DT;
  return r;
}

// ================= K1: propagate + weight + tile sums =================
__global__ void __launch_bounds__(B1)
k_weigh(const float* __restrict__ particles, const float* __restrict__ u,
        const float* __restrict__ z, const float* __restrict__ M,
        const float* __restrict__ eps, const float* __restrict__ Q,
        float* __restrict__ w, float* __restrict__ tileSums) {
  __shared__ float sWr[8];
  const int tid = threadIdx.x;
  const long long i0 = (long long)blockIdx.x * TILE + (long long)tid * PT;

  const float u0 = u[0], u1 = u[1], zz = z[0], q = Q[0];
  const float L00 = sqrtf(M[0]);
  const float L10 = M[2] / L00;
  const float L11 = sqrtf(M[3] - L10 * L10);

  // 4 consecutive particles per thread: 48B + 32B contiguous -> fully coalesced
  const float4* p4 = (const float4*)(particles + i0 * 3);
  float4 pa = p4[0], pb = p4[1], pc = p4[2];
  const float4* e4 = (const float4*)(eps + i0 * 2);
  float4 ea = e4[0], eb = e4[1];

  float px[PT]  = {pa.x, pa.w, pb.z, pc.y};
  float py[PT]  = {pa.y, pb.x, pb.w, pc.z};
  float pth[PT] = {pa.z, pb.y, pc.x, pc.w};
  float e0[PT]  = {ea.x, ea.z, eb.x, eb.z};
  float e1[PT]  = {ea.y, ea.w, eb.y, eb.w};

  float4 wt;
  float* wtp = &wt.x;
  float local = 0.f;
  for (int k = 0; k < PT; ++k) {
    P3 r = propagate(px[k], py[k], pth[k], e0[k], e1[k], u0, u1, L00, L10, L11);
    float zexp  = sqrtf(r.x * r.x + r.y * r.y);
    float innov = zz - zexp;
    float wk = __expf(-0.5f * innov * innov / q) + 1e-8f;
    wtp[k] = wk;
    local += wk;
  }
  *(float4*)(w + i0) = wt;

  // tile sum: WMMA wave reduction + 8-wave LDS combine
  float wsum = wave_reduce_wmma(local);
  int lane = tid & 31, wid = tid >> 5;
  if (lane == 0) sWr[wid] = wsum;
  __syncthreads();
  if (tid == 0) {
    float s = 0.f;
    for (int i = 0; i < B1 / 32; ++i) s += sWr[i];
    tileSums[blockIdx.x] = s;
  }
}

// ================= K2: exclusive scan of tile sums (single block) =================
__global__ void __launch_bounds__(1024)
k_scan_tiles(const float* __restrict__ tileSums, float* __restrict__ tileScan,
             float* __restrict__ accum, int NT) {
  __shared__ float sThr[1024];
  const int tid = threadIdx.x;
  const int R = NT >> 10;          // per-thread count (NT multiple of 1024)
  float v[8];
  float tot = 0.f;
  const int base = tid * R;
  for (int k = 0; k < R; ++k) v[k] = tileSums[base + k];
  for (int k = 0; k < R; ++k) { float t = v[k]; v[k] = tot; tot += t; } // local exclusive
  sThr[tid] = tot;
  __syncthreads();
  // inclusive Hillis-Steele over 1024 thread totals
  for (int off = 1; off < 1024; off <<= 1) {
    float add = (tid >= off) ? sThr[tid - off] : 0.f;
    __syncthreads();
    sThr[tid] += add;
    __syncthreads();
  }
  float excl = sThr[tid] - tot;
  for (int k = 0; k < R; ++k) tileScan[base + k] = excl + v[k];
  if (tid == 1023) tileScan[NT] = sThr[1023];   // total weight W
  if (tid < 4) accum[tid] = 0.f;                // zero moment accumulators
}

// ================= K3: in-place tile-local inclusive scan (w -> c) =================
__global__ void __launch_bounds__(B1)
k_scan_local(float* __restrict__ w, const float* __restrict__ tileScan) {
  __shared__ float sThr[B1];
  const int tid = threadIdx.x;
  const long long i0 = (long long)blockIdx.x * TILE + (long long)tid * PT;
  float4 wv = *(const float4*)(w + i0);
  float v[PT] = {wv.x, wv.y, wv.z, wv.w};
  float tot = v[0] + v[1] + v[2] + v[3];
  sThr[tid] = tot;
  __syncthreads();
  for (int off = 1; off < B1; off <<= 1) {
    float add = (tid >= off) ? sThr[tid - off] : 0.f;
    __syncthreads();
    sThr[tid] += add;
    __syncthreads();
  }
  float run = tileScan[blockIdx.x] + (sThr[tid] - tot);
  float4 cv;
  run += v[0]; cv.x = run;
  run += v[1]; cv.y = run;
  run += v[2]; cv.z = run;
  run += v[3]; cv.w = run;
  *(float4*)(w + i0) = cv;
}

// ================= K4: systematic resample + gather-recompute + moments =================
__global__ void __launch_bounds__(B1)
k_resample(const float* __restrict__ c, const float* __restrict__ tileScan,
           const float* __restrict__ particles, const float* __restrict__ eps,
           const float* __restrict__ u, const float* __restrict__ M,
           const float* __restrict__ r01, float* __restrict__ out,
           float* __restrict__ accum, int N, int NT) {
  __shared__ float sB[MAX_NT + 1];
  __shared__ float sWr[8];
  const int tid = threadIdx.x;

  // Stage tileScan[0..NT-1] into LDS on the CDNA5 async global->LDS path
  // (bypasses the VGPR round-trip; tracked by ASYNCcnt). Inline asm is used
  // because the async builtins' prototypes differ across toolchains.
  {
    const int n128 = NT >> 2;   // NT multiple of 4; tileScan is 16B aligned
    for (int t = tid; t < n128; t += B1) {
      __attribute__((address_space(3))) float* lptr =
          (__attribute__((address_space(3))) float*)(&sB[4 * t]);
      unsigned lds_addr = (unsigned)(unsigned long long)lptr;
      unsigned long long gaddr = (unsigned long long)(tileScan + 4 * t);
      asm volatile("global_load_async_to_lds_b128 %0, %1, off"
                   :
                   : "v"(lds_addr), "v"(gaddr)
                   : "memory");
    }
    asm volatile("s_wait_asynccnt 0" ::: "memory");
  }
  if (tid == 0) sB[NT] = tileScan[NT];
  __syncthreads();

  const float W  = sB[NT];
  const float u0 = u[0], u1 = u[1], r = r01[0];
  const float L00 = sqrtf(M[0]);
  const float L10 = M[2] / L00;
  const float L11 = sqrtf(M[3] - L10 * L10);
  const float scale = W / (float)N;   // search c (unnormalized) for U*W

  const long long j0 = (long long)blockIdx.x * TILE + (long long)tid * PT;
  float sx = 0.f, sy = 0.f, ss = 0.f, sc = 0.f;

  for (int k = 0; k < PT; ++k) {
    const long long j = j0 + k;
    const float T = (r + (float)j) * scale;

    // level 1: lower_bound over tile boundaries sB[t+1] in LDS
    int lo = 0, hi = NT - 1;
    while (lo < hi) {
      int mid = (lo + hi) >> 1;
      if (sB[mid + 1] < T) lo = mid + 1; else hi = mid;
    }
    // level 2: lower_bound inside the 1024-entry tile (L2-resident c)
    int a = lo * TILE, b = a + TILE - 1;
    while (a < b) {
      int mid = (a + b) >> 1;
      if (c[mid] < T) a = mid + 1; else b = mid;
    }
    int idx = a;
    // fixup for WMMA-tree vs sequential-scan rounding at tile seams
    while (idx > 0 && c[idx - 1] >= T) --idx;
    while (idx < N - 1 && c[idx] < T) ++idx;

    // gather source particle and recompute propagation (cheaper than a pbar array)
    float px = particles[3 * idx], py = particles[3 * idx + 1], th = particles[3 * idx + 2];
    float e0 = eps[2 * idx], e1 = eps[2 * idx + 1];
    P3 p = propagate(px, py, th, e0, e1, u0, u1, L00, L10, L11);

    out[3 + 3 * j]     = p.x;
    out[3 + 3 * j + 1] = p.y;
    out[3 + 3 * j + 2] = p.th;

    sx += p.x; sy += p.y;
    float s2, c2;
    __sincosf(p.th, &s2, &c2);
    ss += s2; sc += c2;
  }

  // reduce the 4 moment accumulators on the matrix pipe, then atomics
  float vals[4] = {sx, sy, ss, sc};
  for (int q = 0; q < 4; ++q) {
    float wsum = wave_reduce_wmma(vals[q]);
    int lane = tid & 31, wid = tid >> 5;
    if (lane == 0) sWr[wid] = wsum;
    __syncthreads();
    if (tid == 0) {
      float s = 0.f;
      for (int i = 0; i < B1 / 32; ++i) s += sWr[i];
      atomicAdd(&accum[q], s);
    }
    __syncthreads();
  }
}

// ================= K5: finalize mu =================
__global__ void k_finalize(const float* __restrict__ accum, float* __restrict__ out, int N) {
  if (threadIdx.x == 0 && blockIdx.x == 0) {
    float inv = 1.0f / (float)N;
    out[0] = accum[0] * inv;
    out[1] = accum[1] * inv;
    out[2] = atan2f(accum[2], accum[3]);
  }
}

extern "C" void kernel_launch(void* const* d_in, const int* in_sizes, int n_in,
                              void* d_out, int out_size, void* d_ws, size_t ws_size,
                              hipStream_t stream) {
  (void)n_in; (void)out_size; (void)ws_size;
  const float* particles = (const float*)d_in[0];
  const float* u         = (const float*)d_in[1];
  const float* z         = (const float*)d_in[2];
  const float* M         = (const float*)d_in[3];
  const float* eps       = (const float*)d_in[4];
  const float* r01       = (const float*)d_in[5];
  const float* Q         = (const float*)d_in[6];
  float* out = (float*)d_out;

  const int N  = in_sizes[0] / 3;   // 4,194,304
  const int NT = N / TILE;          // 4096 tiles

  // workspace layout (floats): c[N] | tileSums[NT] | tileScan[NT+1] | pad | accum[4]
  float* ws       = (float*)d_ws;
  float* c        = ws;
  float* tileSums = ws + N;
  float* tileScan = ws + N + NT;
  float* accum    = ws + N + 2 * NT + 2;

  k_weigh     <<<NT, B1,   0, stream>>>(particles, u, z, M, eps, Q, c, tileSums);
  k_scan_tiles<<<1,  1024, 0, stream>>>(tileSums, tileScan, accum, NT);
  k_scan_local<<<NT, B1,   0, stream>>>(c, tileScan);
  k_resample  <<<NT, B1,   0, stream>>>(c, tileScan, particles, eps, u, M, r01,
                                        out, accum, N, NT);
  k_finalize  <<<1,  1,    0, stream>>>(accum, out, N);
}